// DeformableTransformerDecoderLayer_19258633355693
// MI455X (gfx1250) — compile-verified
//
#include <hip/hip_runtime.h>
#include <hip/hip_bf16.h>
#include <math.h>

// ---------------- problem constants ----------------
#define D      256
#define NH     8
#define HD     32     // D / NH
#define NL     4
#define NP     4
#define DFFN   1024
#define NQ     900
#define BS     8
#define LIN    11253
#define ROWS   (NQ * BS)        // 7200
#define VROWS  (LIN * BS)       // 90024
#define EPS    1e-5f

__device__ __constant__ int   c_Hs[NL]     = {92, 46, 23, 12};
__device__ __constant__ int   c_Ws[NL]     = {92, 46, 23, 12};
__device__ __constant__ int   c_starts[NL] = {0, 8464, 10580, 11109};

typedef float v2f __attribute__((ext_vector_type(2)));
typedef float v8f __attribute__((ext_vector_type(8)));

// ---------------------------------------------------------------------------
// Generic GEMM:  C[M,N] = act( (A + A2) @ W^T + bias ),  A:(M,K) W:(N,K)
// fp32 WMMA 16x16x4 (exact fp32 accumulate -> matches the fp32 reference).
//
// Block = 256 threads = 8 waves, computing a 16(M) x 128(N) output strip.
// The 16 A rows are shared by all 8 waves and staged through LDS in
// 64-wide K chunks; the optional addend (tgt+pos) and the ragged-M zeroing
// are applied once at LDS-fill time, so the WMMA inner loop is branchless:
//   ds_load_b64 (A frag) + global_load_b64 (weight frag) + v_wmma per step.
//
// A-frag layout (32b A 16x4): lanes 0-15 rows M0..M15 K={kk,kk+1};
//                             lanes 16-31 same rows   K={kk+2,kk+3}.
// B/C layout: lane%16 = column; C rows r + 8*(lane/16) in v[0..7].
// Requires: N % 128 == 0, K % 64 == 0 (true for all launches here).
// ---------------------------------------------------------------------------
#define KC 64
__global__ __launch_bounds__(256) void gemm_wmma_kernel(
    const float* __restrict__ A, const float* __restrict__ A2,
    const float* __restrict__ W, const float* __restrict__ bias,
    float* __restrict__ C, int M, int N, int K, int relu)
{
    __shared__ float As[16][KC + 2];   // +2 floats: de-conflict banks, keep 8B align

    const int wave = threadIdx.x >> 5;
    const int lane = threadIdx.x & 31;
    const int groupsN = N >> 7;                    // # of 128-wide column groups
    const int mt = blockIdx.x / groupsN;
    const int ng = blockIdx.x - mt * groupsN;
    const int r0 = mt * 16;

    const int half = lane >> 4;
    const int l16  = lane & 15;
    const int bcol = ng * 128 + wave * 16 + l16;
    const float* wptr = W + (size_t)bcol * K + 2 * half;

    v8f c;
    const float bv = bias[bcol];
#pragma unroll
    for (int i = 0; i < 8; ++i) c[i] = bv;

    // LDS-fill indices: 16 rows x 64 k = 1024 elems, 4 per thread
    const int fk    = threadIdx.x & 63;
    const int frow0 = threadIdx.x >> 6;            // 0..3, rows frow0 + 4*i

    for (int k0 = 0; k0 < K; k0 += KC) {
        __syncthreads();
#pragma unroll
        for (int i = 0; i < 4; ++i) {
            const int row  = frow0 + 4 * i;
            const int arow = r0 + row;
            const int arc  = (arow < M) ? arow : (M - 1);
            float vv = A[(size_t)arc * K + k0 + fk];
            if (A2) vv += A2[(size_t)arc * K + k0 + fk];
            if (arow >= M) vv = 0.f;
            As[row][fk] = vv;
        }
        __syncthreads();
#pragma unroll
        for (int kk = 0; kk < KC; kk += 4) {
            v2f a = *(const v2f*)&As[l16][kk + 2 * half];
            v2f b = *(const v2f*)(wptr + k0 + kk);
            c = __builtin_amdgcn_wmma_f32_16x16x4_f32(
                    /*neg_a=*/false, a, /*neg_b=*/false, b,
                    /*c_mod=*/(short)0, c, /*reuse_a=*/false, /*reuse_b=*/false);
        }
    }

#pragma unroll
    for (int r = 0; r < 8; ++r) {
        const int orow = r0 + r + 8 * half;
        if (orow < M) {
            float v = c[r];
            if (relu) v = fmaxf(v, 0.f);
            C[(size_t)orow * N + bcol] = v;
        }
    }
}

// ---------------------------------------------------------------------------
// Fused self-attention with online softmax.
// grid = (BS*NH, ceil(NQ/128)), block = 128. One thread = one query.
// K/V for the (b,h) head are streamed through LDS in 64-key chunks.
// ---------------------------------------------------------------------------
#define ATT_CHUNK 64
__global__ __launch_bounds__(128) void attn_kernel(
    const float* __restrict__ qh, const float* __restrict__ kh,
    const float* __restrict__ vh, float* __restrict__ out, float scale)
{
    const int bh = blockIdx.x;
    const int h = bh % NH;
    const int b = bh / NH;
    const int qi = blockIdx.y * 128 + threadIdx.x;

    __shared__ float Ks[ATT_CHUNK][HD];
    __shared__ float Vs[ATT_CHUNK][HD];

    float qreg[HD];
    if (qi < NQ) {
        const float* qp = qh + ((size_t)qi * BS + b) * D + h * HD;
#pragma unroll
        for (int d = 0; d < HD; ++d) qreg[d] = qp[d] * scale;
    }

    float m = -3.0e38f, ssum = 0.f;
    float acc[HD];
#pragma unroll
    for (int d = 0; d < HD; ++d) acc[d] = 0.f;

    for (int k0 = 0; k0 < NQ; k0 += ATT_CHUNK) {
        const int cnt = min(ATT_CHUNK, NQ - k0);
        __syncthreads();
        for (int t = threadIdx.x; t < cnt * HD; t += 128) {
            const int j = t >> 5, d = t & 31;
            const size_t src = ((size_t)(k0 + j) * BS + b) * D + h * HD + d;
            Ks[j][d] = kh[src];
            Vs[j][d] = vh[src];
        }
        __syncthreads();
        if (qi < NQ) {
            for (int j = 0; j < cnt; ++j) {
                float s = 0.f;
#pragma unroll
                for (int d = 0; d < HD; ++d) s = fmaf(qreg[d], Ks[j][d], s);
                const float mn = fmaxf(m, s);
                const float corr = __expf(m - mn);
                const float e = __expf(s - mn);
                ssum = ssum * corr + e;
#pragma unroll
                for (int d = 0; d < HD; ++d)
                    acc[d] = fmaf(acc[d], corr, e * Vs[j][d]);
                m = mn;
            }
        }
    }
    if (qi < NQ) {
        const float inv = 1.f / ssum;
        float* op = out + ((size_t)qi * BS + b) * D + h * HD;
#pragma unroll
        for (int d = 0; d < HD; ++d) op[d] = acc[d] * inv;
    }
}

// ---------------------------------------------------------------------------
// out = LayerNorm(X + Y) * g + b.  One wave per row of 256.
// ---------------------------------------------------------------------------
__global__ __launch_bounds__(256) void add_ln_kernel(
    const float* __restrict__ X, const float* __restrict__ Y,
    const float* __restrict__ g, const float* __restrict__ beta,
    float* __restrict__ out, int rows)
{
    const int wid = blockIdx.x * 8 + (threadIdx.x >> 5);
    if (wid >= rows) return;
    const int lane = threadIdx.x & 31;
    const float* xr = X + (size_t)wid * D;
    const float* yr = Y + (size_t)wid * D;

    float v[8];
    float s = 0.f, s2 = 0.f;
#pragma unroll
    for (int i = 0; i < 8; ++i) {
        const int d = i * 32 + lane;
        v[i] = xr[d] + yr[d];
        s += v[i];
        s2 = fmaf(v[i], v[i], s2);
    }
#pragma unroll
    for (int off = 16; off >= 1; off >>= 1) {
        s  += __shfl_xor(s,  off, 32);
        s2 += __shfl_xor(s2, off, 32);
    }
    const float mean = s * (1.f / D);
    const float var  = s2 * (1.f / D) - mean * mean;
    const float r = rsqrtf(var + EPS);
    float* orow = out + (size_t)wid * D;
#pragma unroll
    for (int i = 0; i < 8; ++i) {
        const int d = i * 32 + lane;
        orow[d] = (v[i] - mean) * r * g[d] + beta[d];
    }
}

// ---------------------------------------------------------------------------
// MS-deformable attention sampling. One wave per (q,b,h); lane = channel d.
// Includes the softmax over the NL*NP=16 attention logits (computed
// redundantly per lane from L2-cached broadcast loads).
// ---------------------------------------------------------------------------
__global__ __launch_bounds__(256) void deform_kernel(
    const float* __restrict__ value,    // (VROWS, D) rows = i*BS+b
    const float* __restrict__ off,      // (ROWS, 256)
    const float* __restrict__ attnlog,  // (ROWS, 128)
    const float* __restrict__ ref,      // (NQ, BS, NL, 2)
    float* __restrict__ out)            // (ROWS, D)
{
    const int wid = blockIdx.x * 8 + (threadIdx.x >> 5);
    if (wid >= ROWS * NH) return;
    const int lane = threadIdx.x & 31;
    const int h  = wid % NH;
    const int qb = wid / NH;      // == q*BS + b
    const int b  = qb % BS;

    // softmax over the 16 (l,p) logits for this (row, head)
    const float* lg = attnlog + (size_t)qb * (NH * NL * NP) + h * (NL * NP);
    float lv[16], mx = -3.0e38f;
#pragma unroll
    for (int j = 0; j < 16; ++j) { lv[j] = lg[j]; mx = fmaxf(mx, lv[j]); }
    float den = 0.f;
#pragma unroll
    for (int j = 0; j < 16; ++j) { lv[j] = __expf(lv[j] - mx); den += lv[j]; }
    const float inv = 1.f / den;

    float acc = 0.f;
    for (int l = 0; l < NL; ++l) {
        const int Hl = c_Hs[l], Wl = c_Ws[l], start = c_starts[l];
        const float rx = ref[((size_t)qb * NL + l) * 2 + 0];
        const float ry = ref[((size_t)qb * NL + l) * 2 + 1];
        for (int p = 0; p < NP; ++p) {
            const int ch = ((h * NL + l) * NP + p) * 2;
            const float ox = off[(size_t)qb * D + ch];
            const float oy = off[(size_t)qb * D + ch + 1];
            const float lx = fmaf(rx, (float)Wl, ox) - 0.5f;
            const float ly = fmaf(ry, (float)Hl, oy) - 0.5f;
            const float x0f = floorf(lx), y0f = floorf(ly);
            const float fx = lx - x0f, fy = ly - y0f;
            const int x0 = (int)x0f, y0 = (int)y0f;
            const float wgt = lv[l * NP + p] * inv;
#pragma unroll
            for (int corner = 0; corner < 4; ++corner) {
                const int dx = corner & 1, dy = corner >> 1;
                const int xi = x0 + dx, yi = y0 + dy;
                const float cw = (dx ? fx : (1.f - fx)) * (dy ? fy : (1.f - fy));
                const bool valid = (xi >= 0) & (xi < Wl) & (yi >= 0) & (yi < Hl);
                const int xc = min(max(xi, 0), Wl - 1);
                const int yc = min(max(yi, 0), Hl - 1);
                const int idx = start + yc * Wl + xc;
                const float gv = value[((size_t)idx * BS + b) * D + h * HD + lane];
                if (valid) acc = fmaf(wgt * cw, gv, acc);
            }
        }
    }
    out[(size_t)qb * D + h * HD + lane] = acc;
}

// ---------------------------------------------------------------------------
// host side
// ---------------------------------------------------------------------------
static inline void launch_gemm(const float* A, const float* A2, const float* W,
                               const float* bias, float* C,
                               int M, int N, int K, int relu, hipStream_t s)
{
    const int blocks = ((M + 15) / 16) * (N / 128);
    gemm_wmma_kernel<<<blocks, 256, 0, s>>>(A, A2, W, bias, C, M, N, K, relu);
}

extern "C" void kernel_launch(void* const* d_in, const int* in_sizes, int n_in,
                              void* d_out, int out_size, void* d_ws, size_t ws_size,
                              hipStream_t stream)
{
    // ---- inputs (setup_inputs dict order, params flattened in dict order) ----
    const float* tgt    = (const float*)d_in[0];   // (NQ, BS, D)
    const float* pos    = (const float*)d_in[1];   // (NQ, BS, D)
    const float* ref    = (const float*)d_in[2];   // (NQ, BS, NL, 2)
    const float* memory = (const float*)d_in[3];   // (LIN, BS, D)
    const float* wq = (const float*)d_in[4];  const float* bq = (const float*)d_in[5];
    const float* wk = (const float*)d_in[6];  const float* bk = (const float*)d_in[7];
    const float* wv = (const float*)d_in[8];  const float* bv = (const float*)d_in[9];
    const float* wo = (const float*)d_in[10]; const float* bo = (const float*)d_in[11];
    const float* w_off  = (const float*)d_in[12]; const float* b_off  = (const float*)d_in[13];
    const float* w_attn = (const float*)d_in[14]; const float* b_attn = (const float*)d_in[15];
    const float* w_val  = (const float*)d_in[16]; const float* b_val  = (const float*)d_in[17];
    const float* w_out  = (const float*)d_in[18]; const float* b_out  = (const float*)d_in[19];
    const float* n1_g = (const float*)d_in[20]; const float* n1_b = (const float*)d_in[21];
    const float* n2_g = (const float*)d_in[22]; const float* n2_b = (const float*)d_in[23];
    const float* n3_g = (const float*)d_in[24]; const float* n3_b = (const float*)d_in[25];
    const float* w1 = (const float*)d_in[26]; const float* b1 = (const float*)d_in[27];
    const float* w2 = (const float*)d_in[28]; const float* b2 = (const float*)d_in[29];

    // ---- workspace layout ----
    const size_t S = (size_t)ROWS * D;          // 1,843,200 floats
    float* B0  = (float*)d_ws;
    float* B1  = B0 + S;
    float* B2  = B1 + S;
    float* B3  = B2 + S;
    float* B4  = B3 + S;
    float* B5  = B4 + S;
    float* VAL = B5 + S;                        // (VROWS, D)
    float* ATT = VAL + (size_t)VROWS * D;       // (ROWS, 128)
    float* H1  = ATT + (size_t)ROWS * (NH * NL * NP);  // (ROWS, DFFN)
    float* OUT = (float*)d_out;

    const float scale = 1.0f / sqrtf((float)HD);

    // ---- self-attention ----
    launch_gemm(tgt, pos, wq, bq, B0, ROWS, D, D, 0, stream);     // qh
    launch_gemm(tgt, pos, wk, bk, B1, ROWS, D, D, 0, stream);     // kh
    launch_gemm(tgt, nullptr, wv, bv, B2, ROWS, D, D, 0, stream); // vh
    {
        dim3 grid(BS * NH, (NQ + 127) / 128);
        attn_kernel<<<grid, 128, 0, stream>>>(B0, B1, B2, B3, scale);
    }
    launch_gemm(B3, nullptr, wo, bo, B4, ROWS, D, D, 0, stream);  // sa
    add_ln_kernel<<<(ROWS + 7) / 8, 256, 0, stream>>>(tgt, B4, n2_g, n2_b, B5, ROWS); // tgt1

    // ---- MS deformable cross-attention ----
    launch_gemm(memory, nullptr, w_val, b_val, VAL, VROWS, D, D, 0, stream); // value
    launch_gemm(B5, pos, w_off,  b_off,  B0,  ROWS, D, D, 0, stream);        // offsets
    launch_gemm(B5, pos, w_attn, b_attn, ATT, ROWS, NH * NL * NP, D, 0, stream); // attn logits
    deform_kernel<<<(ROWS * NH + 7) / 8, 256, 0, stream>>>(VAL, B0, ATT, ref, B1);
    launch_gemm(B1, nullptr, w_out, b_out, B2, ROWS, D, D, 0, stream);       // ca
    add_ln_kernel<<<(ROWS + 7) / 8, 256, 0, stream>>>(B5, B2, n1_g, n1_b, B3, ROWS); // tgt2

    // ---- FFN ----
    launch_gemm(B3, nullptr, w1, b1, H1, ROWS, DFFN, D, 1, stream);  // relu
    launch_gemm(H1, nullptr, w2, b2, B4, ROWS, D, DFFN, 0, stream);
    add_ln_kernel<<<(ROWS + 7) / 8, 256, 0, stream>>>(B3, B4, n3_g, n3_b, OUT, ROWS);
}